// GENConv_81071802679783
// MI455X (gfx1250) — compile-verified
//
#include <hip/hip_runtime.h>

// GENConv fused pipeline for gfx1250 (MI455X), fp32 end-to-end.
// GEMMs use V_WMMA_F32_16X16X4_F32 (the only fp32 WMMA on CDNA5).

#define N_NODES_C 100000
#define N_EDGES_C 1000000
#define D_EDGE_C  32
#define D_C       64
#define EPS_C     1e-7f

typedef float v2f __attribute__((ext_vector_type(2)));
typedef float v8f __attribute__((ext_vector_type(8)));

// D(16x16,f32) = A(16x4,f32) x B(4x16,f32) + C
// args: (neg_a, A, neg_b, B, c_mod, C, reuse_a, reuse_b)
__device__ __forceinline__ v8f wmma4(v2f a, v2f b, v8f c) {
    return __builtin_amdgcn_wmma_f32_16x16x4_f32(false, a, false, b,
                                                 (short)0, c, false, false);
}

__global__ void zero_kernel(float* __restrict__ p, long long n) {
    long long i = (long long)blockIdx.x * blockDim.x + threadIdx.x;
    if (i < n) p[i] = 0.f;
}

// One wave handles 16 edges x 64 output channels.
// pass==0: m = relu(h_src + e@We + be) + eps ; atomicMax(zmax[dst][c], m)
//          (uint-punned max: all m >= eps > 0, positive floats order as uints)
// pass==1: ez = exp(m - zmax[dst][c]); denom += ez; num += m*ez
__global__ __launch_bounds__(256)
void edge_pass(const float* __restrict__ node_feats,
               const float* __restrict__ edge_feats,
               const int*   __restrict__ src,
               const int*   __restrict__ dst,
               const float* __restrict__ We,
               const float* __restrict__ be,
               float* __restrict__ zmax,
               float* __restrict__ denom,
               float* __restrict__ num,
               int pass)
{
    __shared__ float sWe[D_EDGE_C * D_C];               // 8 KB
    for (int i = threadIdx.x; i < D_EDGE_C * D_C; i += blockDim.x)
        sWe[i] = We[i];
    __syncthreads();

    const int lane = threadIdx.x & 31;
    const int wave = threadIdx.x >> 5;
    const int rA   = lane & 15;   // A-row / D-column selector
    const int half = lane >> 4;   // 0: lanes 0-15, 1: lanes 16-31
    const int edgeBase = (blockIdx.x * 8 + wave) * 16;

    // ---- A fragments: 16x4 f32 tiles of edge_feats over K=32 (8 chunks) ----
    // ISA layout: lanes 0-15 hold {K=k0,k0+1}, lanes 16-31 hold {K=k0+2,k0+3}
    v2f a[8];
    {
        int e  = edgeBase + rA;
        int ec = e < N_EDGES_C ? e : N_EDGES_C - 1;
        const float* ep = edge_feats + (size_t)ec * D_EDGE_C;
        #pragma unroll
        for (int kc = 0; kc < 8; ++kc) {
            int k = kc * 4 + 2 * half;
            v2f t; t.x = ep[k]; t.y = ep[k + 1];
            if (e >= N_EDGES_C) { t.x = 0.f; t.y = 0.f; } // tail: zero A rows
            a[kc] = t;
        }
    }

    // ---- GEMM: acc[nt] = e(16x32) @ We(32x64) tile nt, via 8 WMMAs each ----
    v8f acc[4];
    #pragma unroll
    for (int nt = 0; nt < 4; ++nt) {
        v8f c = {};
        #pragma unroll
        for (int kc = 0; kc < 8; ++kc) {
            int k = kc * 4 + 2 * half;
            v2f b;
            b.x = sWe[(k    ) * D_C + nt * 16 + rA];
            b.y = sWe[(k + 1) * D_C + nt * 16 + rA];
            c = wmma4(a[kc], b, c);
        }
        acc[nt] = c;
    }

    // ---- per-lane edge indices for the 8 D-rows (M = r + 8*half) ----
    int sIdx[8], dIdx[8], valid[8];
    #pragma unroll
    for (int r = 0; r < 8; ++r) {
        int e  = edgeBase + r + 8 * half;
        int ec = e < N_EDGES_C ? e : N_EDGES_C - 1;
        sIdx[r]  = src[ec];
        dIdx[r]  = dst[ec];
        valid[r] = (e < N_EDGES_C);
    }

    // ---- epilogue: gather h_src (L2-resident), relu+eps, segment atomics ----
    #pragma unroll
    for (int nt = 0; nt < 4; ++nt) {
        const int   col  = nt * 16 + rA;
        const float bias = be[col];
        #pragma unroll
        for (int r = 0; r < 8; ++r) {
            if (!valid[r]) continue;
            float m = acc[nt][r] + bias
                    + node_feats[(size_t)sIdx[r] * D_C + col];
            m = (m > 0.f ? m : 0.f) + EPS_C;        // relu + eps; BETA==1 -> z==m
            size_t o = (size_t)dIdx[r] * D_C + col;
            if (pass == 0) {
                atomicMax((unsigned int*)(zmax + o), __float_as_uint(m));
            } else {
                float ez = __expf(m - zmax[o]);
                atomicAdd(denom + o, ez);
                atomicAdd(num   + o, m * ez);
            }
        }
    }
}

// One wave handles 16 nodes x 64 channels: out = (h + num/denom) @ Wm + bm
__global__ __launch_bounds__(256)
void node_pass(const float* __restrict__ node_feats,
               const float* __restrict__ denom,
               const float* __restrict__ num,
               const float* __restrict__ Wm,
               const float* __restrict__ bm,
               float* __restrict__ out)
{
    __shared__ float sWm[D_C * D_C];                    // 16 KB
    for (int i = threadIdx.x; i < D_C * D_C; i += blockDim.x)
        sWm[i] = Wm[i];
    __syncthreads();

    const int lane = threadIdx.x & 31;
    const int wave = threadIdx.x >> 5;
    const int rA   = lane & 15;
    const int half = lane >> 4;
    const int nodeBase = (blockIdx.x * 8 + wave) * 16;

    // ---- A fragments: feats = h + agg, 16 K-chunks of 4, loaded once ----
    v2f a[16];
    {
        int nrow = nodeBase + rA;
        int ncl  = nrow < N_NODES_C ? nrow : N_NODES_C - 1;
        const float* fp = node_feats + (size_t)ncl * D_C;
        const float* dp = denom      + (size_t)ncl * D_C;
        const float* up = num        + (size_t)ncl * D_C;
        #pragma unroll
        for (int kc = 0; kc < 16; ++kc) {
            int k  = kc * 4 + 2 * half;
            float d0 = dp[k], d1 = dp[k + 1];
            v2f t;
            t.x = fp[k]     + (d0 > 0.f ? up[k]     / d0 : 0.f);
            t.y = fp[k + 1] + (d1 > 0.f ? up[k + 1] / d1 : 0.f);
            a[kc] = t;
        }
    }

    v8f acc[4];
    #pragma unroll
    for (int nt = 0; nt < 4; ++nt) {
        v8f c = {};
        #pragma unroll
        for (int kc = 0; kc < 16; ++kc) {
            int k = kc * 4 + 2 * half;
            v2f b;
            b.x = sWm[(k    ) * D_C + nt * 16 + rA];
            b.y = sWm[(k + 1) * D_C + nt * 16 + rA];
            c = wmma4(a[kc], b, c);
        }
        acc[nt] = c;
    }

    #pragma unroll
    for (int nt = 0; nt < 4; ++nt) {
        const int   col  = nt * 16 + rA;
        const float bias = bm[col];
        #pragma unroll
        for (int r = 0; r < 8; ++r) {
            int node = nodeBase + r + 8 * half;
            if (node < N_NODES_C)
                out[(size_t)node * D_C + col] = acc[nt][r] + bias;
        }
    }
}

extern "C" void kernel_launch(void* const* d_in, const int* in_sizes, int n_in,
                              void* d_out, int out_size, void* d_ws, size_t ws_size,
                              hipStream_t stream) {
    (void)in_sizes; (void)n_in; (void)out_size; (void)ws_size;
    const float* node_feats = (const float*)d_in[0];
    const float* edge_feats = (const float*)d_in[1];
    const int*   src        = (const int*)  d_in[2];
    const int*   dst        = (const int*)  d_in[3];
    const float* We         = (const float*)d_in[4];
    const float* be         = (const float*)d_in[5];
    const float* Wm         = (const float*)d_in[6];
    const float* bm         = (const float*)d_in[7];
    float*       out        = (float*)d_out;

    // workspace: zmax | denom | num, each N_NODES x 64 f32 (25.6 MB each)
    float* zmax  = (float*)d_ws;
    float* denom = zmax  + (size_t)N_NODES_C * D_C;
    float* num   = denom + (size_t)N_NODES_C * D_C;

    long long zn = 3LL * N_NODES_C * D_C;
    zero_kernel<<<(int)((zn + 255) / 256), 256, 0, stream>>>(zmax, zn);

    int edgeBlocks = (N_EDGES_C + 127) / 128;   // 16 edges/wave, 8 waves/block
    edge_pass<<<edgeBlocks, 256, 0, stream>>>(node_feats, edge_feats, src, dst,
                                              We, be, zmax, denom, num, 0);
    edge_pass<<<edgeBlocks, 256, 0, stream>>>(node_feats, edge_feats, src, dst,
                                              We, be, zmax, denom, num, 1);

    int nodeBlocks = (N_NODES_C + 127) / 128;   // 16 nodes/wave, 8 waves/block
    node_pass<<<nodeBlocks, 256, 0, stream>>>(node_feats, denom, num, Wm, bm, out);
}